// GraMFormerV2_55980603736511
// MI455X (gfx1250) — compile-verified
//
#include <hip/hip_runtime.h>
#include <hip/hip_bf16.h>

// ---------------------------------------------------------------------------
// GraMFormerV2 forward for MI455X (gfx1250, wave32, WMMA f32 16x16x4)
// B=2, S=512, D=256, N=16, DC=4, DI=512
// ---------------------------------------------------------------------------

typedef float v2f __attribute__((ext_vector_type(2)));
typedef float v8f __attribute__((ext_vector_type(8)));

#define Bn_ 2
#define S_  512
#define D_  256
#define N_  16
#define DI_ 512
#define M_  (Bn_ * S_)     // 1024 rows in all GEMMs
#define E_  (DI_ * N_)     // 8192 state elements per batch

__device__ __forceinline__ v8f wmma4(v2f a, v2f b, v8f c) {
  // D = A(16x4, f32) * B(4x16, f32) + C(16x16, f32)  -> v_wmma_f32_16x16x4_f32
  return __builtin_amdgcn_wmma_f32_16x16x4_f32(
      /*neg_a=*/false, a, /*neg_b=*/false, b,
      /*c_mod=*/(short)0, c, /*reuse_a=*/false, /*reuse_b=*/false);
}

__device__ __forceinline__ float sigmoidf_(float x) {
  return 1.f / (1.f + __expf(-x));
}

// ---------------------------------------------------------------------------
// 1) LayerNorm: one block (256 threads) per row of x (B*S rows, D=256)
// ---------------------------------------------------------------------------
__global__ __launch_bounds__(256) void k_layernorm(
    const float* __restrict__ x, const float* __restrict__ ln_w,
    const float* __restrict__ ln_b, float* __restrict__ xn) {
  int row = blockIdx.x, tid = threadIdx.x;
  __shared__ float s1[256], s2[256];
  float v = x[(size_t)row * D_ + tid];
  s1[tid] = v; s2[tid] = v * v;
  __syncthreads();
  for (int off = 128; off > 0; off >>= 1) {
    if (tid < off) { s1[tid] += s1[tid + off]; s2[tid] += s2[tid + off]; }
    __syncthreads();
  }
  float mu  = s1[0] * (1.f / D_);
  float var = s2[0] * (1.f / D_) - mu * mu;
  float r   = rsqrtf(var + 1e-5f);
  xn[(size_t)row * D_ + tid] = (v - mu) * r * ln_w[tid] + ln_b[tid];
}

// ---------------------------------------------------------------------------
// 2) xz = xn @ W_in.T   (M=1024, N=1024, K=256), WMMA f32 16x16x4
//    8 waves/block, one 16x16 tile per wave. 4096 tiles -> 512 blocks.
// ---------------------------------------------------------------------------
__global__ __launch_bounds__(256) void k_gemm_win(
    const float* __restrict__ xn, const float* __restrict__ Win,
    float* __restrict__ xz) {
  const int N = 1024, K = D_;
  int wave = threadIdx.x >> 5, lane = threadIdx.x & 31;
  int tile = blockIdx.x * 8 + wave;
  int tm = (tile / (N / 16)) * 16;
  int tn = (tile % (N / 16)) * 16;
  int mrow = tm + (lane & 15);
  int ncol = tn + (lane & 15);
  int kb   = (lane >> 4) << 1;
  v8f acc = {};
  for (int k0 = 0; k0 < K; k0 += 4) {
    v2f a, b;
    const float* ap = xn  + (size_t)mrow * K + (k0 + kb);
    const float* bp = Win + (size_t)ncol * K + (k0 + kb);   // B[k][n] = Win[n][k]
    a.x = ap[0]; a.y = ap[1];
    b.x = bp[0]; b.y = bp[1];
    acc = wmma4(a, b, acc);
  }
  int n = lane & 15, mo = (lane >> 4) * 8;
#pragma unroll
  for (int r = 0; r < 8; ++r)
    xz[(size_t)(tm + mo + r) * N + tn + n] = acc[r];
}

// ---------------------------------------------------------------------------
// 3) Causal conv as 4 accumulated GEMMs + bias + SiLU.
//    xc[b,t,o] = sum_{k=0..3} sum_i xp[b,t-3+k,i] * conv_w[o,i,k]
//    M=1024, N=512, K=512 per tap. 2048 tiles -> 256 blocks.
// ---------------------------------------------------------------------------
__global__ __launch_bounds__(256) void k_conv_silu(
    const float* __restrict__ xz, const float* __restrict__ conv_w,
    const float* __restrict__ conv_b, float* __restrict__ x_act) {
  const int N = DI_, K = DI_, LDXZ = 1024;
  int wave = threadIdx.x >> 5, lane = threadIdx.x & 31;
  int tile = blockIdx.x * 8 + wave;
  int tm = (tile / (N / 16)) * 16;
  int tn = (tile % (N / 16)) * 16;
  int mrow = tm + (lane & 15);
  int ncol = tn + (lane & 15);
  int kb   = (lane >> 4) << 1;
  int bidx = mrow >> 9;      // batch
  int t    = mrow & (S_ - 1);
  v8f acc = {};
  for (int ktap = 0; ktap < 4; ++ktap) {
    int ts = t - 3 + ktap;
    bool valid = (ts >= 0);
    const float* ap0 = xz + ((size_t)(bidx << 9) + (valid ? ts : 0)) * LDXZ; // xp half
    for (int k0 = 0; k0 < K; k0 += 4) {
      v2f a, b;
      a.x = valid ? ap0[k0 + kb]     : 0.f;
      a.y = valid ? ap0[k0 + kb + 1] : 0.f;
      // conv_w (O=DI, I=DI, H=4): element (o,i,k) at o*2048 + i*4 + k
      const float* bp = conv_w + (((size_t)ncol * DI_ + (k0 + kb)) << 2) + ktap;
      b.x = bp[0]; b.y = bp[4];
      acc = wmma4(a, b, acc);
    }
  }
  int n = lane & 15, mo = (lane >> 4) * 8;
  float cb = conv_b[tn + n];
#pragma unroll
  for (int r = 0; r < 8; ++r) {
    float v = acc[r] + cb;
    x_act[(size_t)(tm + mo + r) * N + tn + n] = v * sigmoidf_(v);
  }
}

// ---------------------------------------------------------------------------
// 4) ssm = x_act @ W_xp.T  (M=1024, N=33 padded to 48, K=512). 192 tiles -> 24 blocks.
//    Layout in ws: ssm[m][48]: [0]=s0, [1..16]=Bm, [17..32]=Cm, [33..47]=0.
// ---------------------------------------------------------------------------
__global__ __launch_bounds__(256) void k_gemm_ssm(
    const float* __restrict__ x_act, const float* __restrict__ Wxp,
    float* __restrict__ ssm) {
  const int NPAD = 48, K = DI_;
  int wave = threadIdx.x >> 5, lane = threadIdx.x & 31;
  int tile = blockIdx.x * 8 + wave;
  int tm = (tile / 3) * 16;
  int tn = (tile % 3) * 16;
  int mrow = tm + (lane & 15);
  int ncol = tn + (lane & 15);
  int kb   = (lane >> 4) << 1;
  bool nvalid = (ncol < 33);
  v8f acc = {};
  for (int k0 = 0; k0 < K; k0 += 4) {
    v2f a, b;
    const float* ap = x_act + (size_t)mrow * K + (k0 + kb);
    a.x = ap[0]; a.y = ap[1];
    const float* bp = Wxp + (size_t)ncol * K + (k0 + kb);
    b.x = nvalid ? bp[0] : 0.f;
    b.y = nvalid ? bp[1] : 0.f;
    acc = wmma4(a, b, acc);
  }
  int n = lane & 15, mo = (lane >> 4) * 8;
#pragma unroll
  for (int r = 0; r < 8; ++r)
    ssm[(size_t)(tm + mo + r) * NPAD + tn + n] = acc[r];
}

// ---------------------------------------------------------------------------
// 5) Sequential scan, sharded across state elements.
//    The recurrence is elementwise-separable in e=(d,n): g[e] needs only
//    hist[:,e]; the boost needs only g-row d; y[d] needs only h-row d.
//    => 64 workgroups x 256 threads, 1 element per thread, 16 d-rows per WG.
//    Only intra-WG barriers are needed per step; no cross-WG communication.
//    Each WG redundantly stages the adjacency row + wsum (L2-resident).
// ---------------------------------------------------------------------------
#define SCAN_TPB 256
#define SCAN_WG_PER_B (E_ / SCAN_TPB)   // 32 WGs per batch, 64 total

__global__ __launch_bounds__(SCAN_TPB) void k_scan(
    const float* __restrict__ adj, const float* __restrict__ x_act,
    const float* __restrict__ ssm, const float* __restrict__ W_dt,
    const float* __restrict__ b_dt, const float* __restrict__ Wr,
    const float* __restrict__ br, float* __restrict__ hist,
    float* __restrict__ y) {
  int b   = blockIdx.x / SCAN_WG_PER_B;
  int e0  = (blockIdx.x % SCAN_WG_PER_B) * SCAN_TPB;
  int tid = threadIdx.x;
  int e   = e0 + tid;             // global state element owned by this thread
  int d   = e >> 4, n = e & 15;
  int d0  = e0 >> 4;              // first d-row owned by this WG (16 rows)

  __shared__ float w_lds[S_];     // masked adjacency row
  __shared__ float s1[SCAN_TPB];  // reduction scratch
  __shared__ float g_lds[SCAN_TPB];
  __shared__ float p_lds[SCAN_TPB];

  float h = 0.f;
  float wdt = W_dt[d], bdt = b_dt[d];
  float wr_row[16];
#pragma unroll
  for (int n2 = 0; n2 < 16; ++n2) wr_row[n2] = Wr[n * 16 + n2];
  float brn = br[n];

  const float* adj_b = adj + (size_t)b * S_ * S_;
  const float* hist_b = hist + (size_t)b * S_ * E_;   // read view
  float* hist_w = hist + (size_t)b * S_ * E_;

  for (int t = 0; t < S_; ++t) {
    // --- stage masked adjacency row w[s] = adj[b,t,s] * (s < t)
    {
      int s_a = tid, s_b2 = tid + 256;
      w_lds[s_a]  = (s_a  < t) ? adj_b[(size_t)t * S_ + s_a]  : 0.f;
      w_lds[s_b2] = (s_b2 < t) ? adj_b[(size_t)t * S_ + s_b2] : 0.f;
    }
    __syncthreads();
    // --- wsum reduction (w >= 0, so wsum>0 <=> any(w>0))
    s1[tid] = w_lds[tid] + w_lds[tid + 256];
    __syncthreads();
    for (int off = 128; off > 0; off >>= 1) {
      if (tid < off) s1[tid] += s1[tid + off];
      __syncthreads();
    }
    float wsum   = s1[0];
    float invdeg = 1.f / fmaxf(wsum, 1.f);
    bool doBoost = (t > 0) && (wsum > 0.f);
    __syncthreads();    // s1 free again

    // --- g[e] = (sum_{s<t} w[s] * hist[s,e]) / deg  (coalesced 1KB/step slab)
    float acc = 0.f;
    if (doBoost) {
      for (int s = 0; s < t; ++s) {
        if (s + 1 < t)
          __builtin_prefetch(hist_b + (size_t)(s + 1) * E_ + e, 0, 0);
        acc += w_lds[s] * hist_b[(size_t)s * E_ + e];
      }
    }
    g_lds[tid] = acc * invdeg;
    __syncthreads();

    // --- state update + boost
    const float* ssm_t = ssm + (size_t)(b * S_ + t) * 48;
    float s0  = ssm_t[0];
    float arg = s0 * wdt + bdt;
    float sp  = (arg > 20.f) ? arg : __logf(1.f + __expf(arg));  // softplus
    float hv  = h * __expf(-sp) + x_act[(size_t)(b * S_ + t) * DI_ + d] * ssm_t[1 + n];
    if (doBoost) {
      float gb = brn;
      const float* grow = g_lds + ((d - d0) << 4);
#pragma unroll
      for (int n2 = 0; n2 < 16; ++n2) gb += grow[n2] * wr_row[n2];
      hv += 0.1f * gb * sigmoidf_(gb);
    }
    h = hv;
    hist_w[(size_t)t * E_ + e] = hv;
    p_lds[tid] = hv * ssm_t[17 + n];   // h * C_t[n]
    __syncthreads();

    // --- y[b,t,d] = sum_n h[d,n]*C_t[n]  (16 rows per WG)
    if (tid < 16) {
      float s = 0.f;
      const float* pr = p_lds + (tid << 4);
#pragma unroll
      for (int n2 = 0; n2 < 16; ++n2) s += pr[n2];
      y[(size_t)(b * S_ + t) * DI_ + d0 + tid] = s;
    }
    __syncthreads();   // protect w_lds/p_lds before next step
  }
}

// ---------------------------------------------------------------------------
// 6a) u = y * silu(z)
// ---------------------------------------------------------------------------
__global__ __launch_bounds__(256) void k_mul_silu(
    const float* __restrict__ y, const float* __restrict__ xz,
    float* __restrict__ u) {
  int i = blockIdx.x * blockDim.x + threadIdx.x;   // 1024*512 elements
  int m = i >> 9, k = i & (DI_ - 1);
  float z = xz[(size_t)m * 1024 + DI_ + k];
  u[i] = y[i] * z * sigmoidf_(z);
}

// ---------------------------------------------------------------------------
// 6b) out = u @ W_out.T + residual  (M=1024, N=256, K=512). 1024 tiles -> 128 blocks.
// ---------------------------------------------------------------------------
__global__ __launch_bounds__(256) void k_gemm_out(
    const float* __restrict__ u, const float* __restrict__ Wout,
    const float* __restrict__ x, float* __restrict__ out) {
  const int N = D_, K = DI_;
  int wave = threadIdx.x >> 5, lane = threadIdx.x & 31;
  int tile = blockIdx.x * 8 + wave;
  int tm = (tile / (N / 16)) * 16;
  int tn = (tile % (N / 16)) * 16;
  int mrow = tm + (lane & 15);
  int ncol = tn + (lane & 15);
  int kb   = (lane >> 4) << 1;
  v8f acc = {};
  for (int k0 = 0; k0 < K; k0 += 4) {
    v2f a, b;
    const float* ap = u    + (size_t)mrow * K + (k0 + kb);
    const float* bp = Wout + (size_t)ncol * K + (k0 + kb);
    a.x = ap[0]; a.y = ap[1];
    b.x = bp[0]; b.y = bp[1];
    acc = wmma4(a, b, acc);
  }
  int n = lane & 15, mo = (lane >> 4) * 8;
#pragma unroll
  for (int r = 0; r < 8; ++r) {
    size_t idx = (size_t)(tm + mo + r) * N + tn + n;
    out[idx] = acc[r] + x[idx];
  }
}

// ---------------------------------------------------------------------------
// Host launcher
// ---------------------------------------------------------------------------
extern "C" void kernel_launch(void* const* d_in, const int* in_sizes, int n_in,
                              void* d_out, int out_size, void* d_ws, size_t ws_size,
                              hipStream_t stream) {
  (void)in_sizes; (void)n_in; (void)out_size; (void)ws_size;
  const float* x      = (const float*)d_in[0];
  const float* adj    = (const float*)d_in[1];
  const float* ln_w   = (const float*)d_in[2];
  const float* ln_b   = (const float*)d_in[3];
  const float* W_in   = (const float*)d_in[4];
  const float* conv_w = (const float*)d_in[5];
  const float* conv_b = (const float*)d_in[6];
  const float* W_xp   = (const float*)d_in[7];
  const float* W_dt   = (const float*)d_in[8];
  const float* b_dt   = (const float*)d_in[9];
  const float* Wr     = (const float*)d_in[10];
  const float* br     = (const float*)d_in[11];
  const float* W_out  = (const float*)d_in[12];
  float* out = (float*)d_out;

  // Workspace layout (floats): ~45.3 MB total
  float* ws    = (float*)d_ws;
  float* xn    = ws;                       // 1024*256   =   262144
  float* xz    = xn    + 262144;           // 1024*1024  =  1048576
  float* x_act = xz    + 1048576;          // 1024*512   =   524288
  float* ssm   = x_act + 524288;           // 1024*48    =    49152
  float* yb    = ssm   + 49152;            // 1024*512   =   524288
  float* ub    = yb    + 524288;           // 1024*512   =   524288
  float* hist  = ub    + 524288;           // 2*512*8192 =  8388608

  k_layernorm<<<M_, 256, 0, stream>>>(x, ln_w, ln_b, xn);
  k_gemm_win <<<512, 256, 0, stream>>>(xn, W_in, xz);
  k_conv_silu<<<256, 256, 0, stream>>>(xz, conv_w, conv_b, x_act);
  k_gemm_ssm <<<24, 256, 0, stream>>>(x_act, W_xp, ssm);
  k_scan     <<<Bn_ * SCAN_WG_PER_B, SCAN_TPB, 0, stream>>>(
      adj, x_act, ssm, W_dt, b_dt, Wr, br, hist, yb);
  k_mul_silu <<<2048, 256, 0, stream>>>(yb, xz, ub);
  k_gemm_out <<<128, 256, 0, stream>>>(ub, W_out, x, out);
}